// NoParamRNNCell_31121333026897
// MI455X (gfx1250) — compile-verified
//
#include <hip/hip_runtime.h>

#ifndef __has_builtin
#define __has_builtin(x) 0
#endif

// gfx1250 hardware tanh (TRANS op). Falls back to libm tanhf if the builtin
// is missing on this toolchain.
__device__ __forceinline__ float htanh(float v) {
#if __has_builtin(__builtin_amdgcn_tanhf)
  return __builtin_amdgcn_tanhf(v);
#else
  return tanhf(v);
#endif
}

static constexpr int B   = 32;
static constexpr int T   = 2048;
static constexpr int U   = 256;
static constexpr int PF  = 8;    // software-pipeline depth in time steps
static constexpr int BPB = 4;    // b-rows per 256-thread block
static constexpr int DPF = 64;   // prefetch distance in time steps

// One recurrence step, minimized dependent chain:
//   h' = tanh( fma(0.5, tanh((0.5*x)*h), h + 0.5) )
// Critical path on h: v_mul -> v_tanh -> v_fma -> v_tanh (2 TRANS, 2 VALU).
// (0.5*x) is off-chain; (h+0.5) is on a parallel branch.
__device__ __forceinline__ float rnn_step(float xhalf, float h) {
  float m  = xhalf * h;                       // on-chain mul
  float t1 = htanh(m);                        // v_tanh_f32
  float s  = __builtin_fmaf(0.5f, t1, h + 0.5f);
  return htanh(s);                            // v_tanh_f32
}

__global__ __launch_bounds__(256, 1)
void rnn_scan_kernel(const float* __restrict__ x, float* __restrict__ out) {
  const int ug = threadIdx.x & 63;   // u-group: 4 consecutive u (one float4)
  const int bl = threadIdx.x >> 6;   // b-row within block
  const int b  = blockIdx.x * BPB + bl;

  const size_t base = ((size_t)b * T) * (size_t)U + (size_t)ug * 4;
  const float4* __restrict__ xp = reinterpret_cast<const float4*>(x + base);
  float4* __restrict__ op       = reinterpret_cast<float4*>(out + base);
  constexpr int strideT = U / 4;     // float4 stride between time steps

  float4 h = make_float4(0.f, 0.f, 0.f, 0.f);

  // Prologue: fill the register ring buffer with steps 0..PF-1,
  // pre-scaled by 0.5 (off the recurrence chain).
  float4 buf[PF];
#pragma unroll
  for (int i = 0; i < PF; ++i) {
    float4 v = xp[(size_t)i * strideT];
    buf[i] = make_float4(0.5f * v.x, 0.5f * v.y, 0.5f * v.z, 0.5f * v.w);
  }

  for (int tc = 0; tc < T; tc += PF) {
    // Long-range prefetch to warm the cache hierarchy well ahead of the
    // register pipeline (lowers to global_prefetch_b8 on gfx1250).
    if (tc + DPF < T) {
      __builtin_prefetch(reinterpret_cast<const void*>(xp + (size_t)(tc + DPF) * strideT), 0, 1);
    }

    // Issue next chunk's loads before computing the current chunk.
    float4 nxt[PF];
    const int tn = tc + PF;
    if (tn < T) {
#pragma unroll
      for (int i = 0; i < PF; ++i) {
        nxt[i] = xp[(size_t)(tn + i) * strideT];
      }
    } else {
#pragma unroll
      for (int i = 0; i < PF; ++i) nxt[i] = make_float4(0.f, 0.f, 0.f, 0.f);
    }

    // Compute + store the current chunk (stores never gate the chain).
#pragma unroll
    for (int i = 0; i < PF; ++i) {
      const float4 xh = buf[i];
      h.x = rnn_step(xh.x, h.x);
      h.y = rnn_step(xh.y, h.y);
      h.z = rnn_step(xh.z, h.z);
      h.w = rnn_step(xh.w, h.w);
      op[(size_t)(tc + i) * strideT] = h;
    }

    // Rotate: pre-scale the freshly loaded chunk (off-chain VALU).
#pragma unroll
    for (int i = 0; i < PF; ++i) {
      buf[i] = make_float4(0.5f * nxt[i].x, 0.5f * nxt[i].y,
                           0.5f * nxt[i].z, 0.5f * nxt[i].w);
    }
  }
}

extern "C" void kernel_launch(void* const* d_in, const int* in_sizes, int n_in,
                              void* d_out, int out_size, void* d_ws, size_t ws_size,
                              hipStream_t stream) {
  (void)in_sizes; (void)n_in; (void)d_ws; (void)ws_size; (void)out_size;
  const float* x = reinterpret_cast<const float*>(d_in[0]);
  float* out     = reinterpret_cast<float*>(d_out);

  dim3 grid(B / BPB);   // 8 blocks
  dim3 block(256);      // 64 u-groups x 4 b-rows, 8 waves (wave32)
  rnn_scan_kernel<<<grid, block, 0, stream>>>(x, out);
}